// AttentionLSTM_91036126806922
// MI455X (gfx1250) — compile-verified
//
#include <hip/hip_runtime.h>
#include <cmath>

typedef __attribute__((ext_vector_type(2))) float v2f;
typedef __attribute__((ext_vector_type(8))) float v8f;

#if defined(__has_builtin)
#  if __has_builtin(__builtin_amdgcn_tensor_load_to_lds)
#    if __clang_major__ >= 23
#      define USE_TDM 2   /* 6-arg builtin (amdgpu-toolchain / clang-23+) */
#    else
#      define USE_TDM 1   /* 5-arg builtin (ROCm 7.2 / clang-22) */
#    endif
#  endif
#  if __has_builtin(__builtin_amdgcn_s_wait_tensorcnt)
#    define HAVE_WAIT_TENSORCNT 1
#  endif
#endif
#ifndef USE_TDM
#  define USE_TDM 0
#endif
#ifndef HAVE_WAIT_TENSORCNT
#  define HAVE_WAIT_TENSORCNT 0
#endif

constexpr int N_ = 128, T_ = 256, D_ = 1024, H_ = 1024, P_ = 196;
constexpr float SCALE_ = 0.03125f; // 1/sqrt(H)

__device__ __forceinline__ float wave_sum(float v) {
  #pragma unroll
  for (int off = 16; off > 0; off >>= 1) v += __shfl_xor(v, off, 32);
  return v;
}

#if USE_TDM
__device__ __forceinline__ void wait_tensorcnt_le(int n) {
#if HAVE_WAIT_TENSORCNT
  if (n == 0) __builtin_amdgcn_s_wait_tensorcnt(0);
  else        __builtin_amdgcn_s_wait_tensorcnt(1);
#else
  if (n == 0) asm volatile("s_wait_tensorcnt 0x0" ::: "memory");
  else        asm volatile("s_wait_tensorcnt 0x1" ::: "memory");
#endif
}
#endif

// ---------------------------------------------------------------------------
// Kernel 1: h0 = c0 = mean over P of A_flat.  One wave per 8 rows, coalesced.
// ---------------------------------------------------------------------------
__global__ __launch_bounds__(256) void init_pool_kernel(
    const float* __restrict__ A, float* __restrict__ h0, float* __restrict__ c0) {
  int lane = threadIdx.x & 31, w = threadIdx.x >> 5;
  int row0 = blockIdx.x * 64 + w * 8;
  for (int r = 0; r < 8; ++r) {
    size_t row = (size_t)(row0 + r);
    const float* rp = A + row * (size_t)P_;
    float acc = 0.f;
    #pragma unroll
    for (int i = 0; i < 7; ++i) { int p = lane + 32 * i; if (p < P_) acc += rp[p]; }
    acc = wave_sum(acc);
    if (lane == 0) { float m = acc * (1.0f / (float)P_); h0[row] = m; c0[row] = m; }
  }
}

// ---------------------------------------------------------------------------
// Kernel 2 (per step): attention over spatial positions for one n per block.
//   scores[p] = sum_h h[h]*A[n,h,p] * scale ; softmax ; attn[h] = sum_p A*w
// Both passes read A rows contiguously (coalesced); A stays L2-resident.
// ---------------------------------------------------------------------------
__global__ __launch_bounds__(256) void attn_kernel(
    const float* __restrict__ A, const float* __restrict__ hprev,
    float* __restrict__ attn) {
  __shared__ float h_sh[H_];
  __shared__ float ps[8][200];
  __shared__ float w_sh[P_];
  __shared__ float red[256];
  int tid = threadIdx.x, lane = tid & 31, w = tid >> 5;
  int n = blockIdx.x;
  const float* An = A + (size_t)n * H_ * P_;

  for (int i = tid; i < H_; i += 256) h_sh[i] = hprev[(size_t)n * H_ + i];
  __syncthreads();

  float sc[7];
  #pragma unroll
  for (int i = 0; i < 7; ++i) sc[i] = 0.f;
  for (int h = w * 128; h < w * 128 + 128; ++h) {
    float hv = h_sh[h];
    const float* row = An + (size_t)h * P_;
    #pragma unroll
    for (int i = 0; i < 7; ++i) { int p = lane + 32 * i; if (p < P_) sc[i] = fmaf(hv, row[p], sc[i]); }
  }
  #pragma unroll
  for (int i = 0; i < 7; ++i) { int p = lane + 32 * i; if (p < P_) ps[w][p] = sc[i]; }
  __syncthreads();

  float s = -__builtin_inff();
  if (tid < P_) {
    s = 0.f;
    #pragma unroll
    for (int wv = 0; wv < 8; ++wv) s += ps[wv][tid];
    s *= SCALE_;
  }
  red[tid] = s;
  __syncthreads();
  for (int off = 128; off > 0; off >>= 1) {
    if (tid < off) red[tid] = fmaxf(red[tid], red[tid + off]);
    __syncthreads();
  }
  float mx = red[0];
  __syncthreads();
  float e = (tid < P_) ? __expf(s - mx) : 0.f;
  red[tid] = e;
  __syncthreads();
  for (int off = 128; off > 0; off >>= 1) {
    if (tid < off) red[tid] += red[tid + off];
    __syncthreads();
  }
  float inv = 1.f / red[0];
  __syncthreads();
  if (tid < P_) w_sh[tid] = e * inv;
  __syncthreads();

  for (int h = w * 128; h < w * 128 + 128; ++h) {
    const float* row = An + (size_t)h * P_;
    float acc = 0.f;
    #pragma unroll
    for (int i = 0; i < 7; ++i) { int p = lane + 32 * i; if (p < P_) acc = fmaf(row[p], w_sh[p], acc); }
    acc = wave_sum(acc);
    if (lane == 0) attn[(size_t)n * H_ + h] = acc;
  }
}

// ---------------------------------------------------------------------------
// Tensor Data Mover: DMA one 32(K-rows) x 4(gate-chunks) x 32(cols) f32 tile
// of weights into LDS with 16-DWORD padding after every 128 DWORDs, i.e.
// destination row stride 144 floats == B_sh layout.  3D descriptor:
//   dim0 = 32 contiguous floats, dim1 = 4 chunks (stride 1024),
//   dim2 = 32 K-rows (stride 4096).
// ---------------------------------------------------------------------------
#if USE_TDM
typedef unsigned int v4u __attribute__((ext_vector_type(4)));
typedef int v8i __attribute__((ext_vector_type(8)));
typedef int v4i __attribute__((ext_vector_type(4)));

__device__ __forceinline__ void tdm_load_b_tile(const float* gsrc, unsigned lds_off) {
  unsigned long long ga = (unsigned long long)(const void*)gsrc;
  v4u g0;
  g0[0] = 1u;                                               // count=1, user mode
  g0[1] = lds_off;                                          // lds_addr (bytes)
  g0[2] = (unsigned)(ga & 0xFFFFFFFFu);                     // global_addr[31:0]
  g0[3] = (unsigned)((ga >> 32) & 0x1FFFFFFu) | (2u << 30); // addr[56:32] | type=2
  v8i g1;
  g1[0] = (int)((2u << 16) |      // data_size = 4 bytes
                (1u << 20) |      // pad_enable
                (6u << 22) |      // pad_interval: every 128 DWORDs
                (15u << 25));     // pad_amount: 16 DWORDs
  g1[1] = (int)(32u << 16);       // tensor_dim0 = 32 (bits 79:48 low half)
  g1[2] = (int)(4u << 16);        // tensor_dim1 = 4  (bits 111:80 low half)
  g1[3] = (int)(32u << 16);       // tile_dim0 = 32   (bits 127:112)
  g1[4] = (int)((32u << 16) | 4u);// tile_dim2 = 32 | tile_dim1 = 4
  g1[5] = 1024;                   // tensor_dim0_stride (low 32)
  g1[6] = (int)(4096u << 16);     // tensor_dim1_stride low16 in [31:16]
  g1[7] = 0;                      // tensor_dim1_stride high
  v4i g2;
  g2[0] = 32;                     // tensor_dim2 = 32
  g2[1] = 0;                      // tensor_dim3 (unused)
  g2[2] = 4096;                   // tensor_dim2_stride (low 32)
  g2[3] = 0;                      // stride hi | tile_dim3 = 0
  v4i g3 = {0, 0, 0, 0};
#if USE_TDM == 1
  __builtin_amdgcn_tensor_load_to_lds(g0, g1, g2, g3, 0);
#else
  v8i gx = {0, 0, 0, 0, 0, 0, 0, 0};
  __builtin_amdgcn_tensor_load_to_lds(g0, g1, g2, g3, gx, 0);
#endif
}
#endif

// ---------------------------------------------------------------------------
// Kernel 3 (per step): fused gate GEMM + LSTM update, WMMA f32 16x16x4.
// Block: 16 batch rows x (32 cols in EACH of the 4 gate chunks) = 16x128 outs.
// K = 3072 (xt | h | attn), streamed in 32-wide chunks through LDS; B tiles
// double-buffered and (when available) DMAed by the Tensor Data Mover.
// ---------------------------------------------------------------------------
__global__ __launch_bounds__(256) void gates_kernel(
    const float* __restrict__ x, const float* __restrict__ Wx,
    const float* __restrict__ Wh, const float* __restrict__ Wattn,
    const float* __restrict__ bias, const float* __restrict__ attn,
    const float* __restrict__ hprev, float* __restrict__ hnext,
    float* __restrict__ cstate, float* __restrict__ out, int t) {
  constexpr int BS = 144;  // B_sh row stride (128 + 16 pad)
  constexpr int AS = 33;   // act_sh row stride
  constexpr int OS = 132;  // a_sh row stride
  __shared__ float B_sh[2][32 * BS];
  __shared__ float act_sh[16 * AS];
  __shared__ float a_sh[16 * OS];

  int tid = threadIdx.x;
  int lane = tid & 31, w = tid >> 5;
  int mb = blockIdx.x >> 5;  // 0..7  (16-row M tile)
  int cg = blockIdx.x & 31;  // 0..31 (32-col group within each 1024 gate chunk)
  int mbase = mb * 16;

  int ar = lane & 15;
  int khalf = (lane >> 4) << 1;                  // 0 or 2 (K half per lane group)
  int colb = ((w >> 1) << 5) + ((w & 1) << 4);   // LDS col base of this wave's tile

#if !USE_TDM
  // per-thread global B-load coordinates (fully coalesced fallback path)
  int bkk = tid >> 7;        // 0/1
  int bcc = tid & 127;       // col within 128
  int bq = bcc >> 5, bc = bcc & 31;
  size_t bcol = (size_t)bq * 1024 + (size_t)cg * 32 + bc;
#endif

  v8f acc = {};

#if USE_TDM
  unsigned lds_b0 = (unsigned)(unsigned long long)
      (__attribute__((address_space(3))) float*)&B_sh[0][0];
  unsigned lds_b1 = (unsigned)(unsigned long long)
      (__attribute__((address_space(3))) float*)&B_sh[1][0];
  if (w == 0) {
    // prologue: DMA chunk 0 (segment 0 -> Wx) into buffer 0
    tdm_load_b_tile(Wx + (size_t)cg * 32, lds_b0);
  }
#endif

  for (int kc = 0; kc < 96; ++kc) {
    int k0 = kc * 32;
    int seg = k0 >> 10;          // 0: x/Wx  1: h/Wh  2: attn/Wattn
    int kl0 = k0 & 1023;
    float* Bcur = &B_sh[kc & 1][0];

    __syncthreads();  // all waves done reading previous buffers / act_sh

#if USE_TDM
    if (w == 0 && kc + 1 < 96) {
      int k1 = k0 + 32; int seg1 = k1 >> 10; int kl1 = k1 & 1023;
      const float* Wn = (seg1 == 0) ? Wx : ((seg1 == 1) ? Wh : Wattn);
      unsigned dst = ((kc + 1) & 1) ? lds_b1 : lds_b0;
      tdm_load_b_tile(Wn + (size_t)kl1 * 4096 + (size_t)cg * 32, dst);
    }
#else
    {
      const float* W = (seg == 0) ? Wx : ((seg == 1) ? Wh : Wattn);
      if (kc + 1 < 96) {  // prefetch next weight chunk into cache
        int k1 = k0 + 32; int seg1 = k1 >> 10; int kl1 = k1 & 1023;
        const float* Wn = (seg1 == 0) ? Wx : ((seg1 == 1) ? Wh : Wattn);
        __builtin_prefetch(&Wn[(size_t)(kl1 + bkk) * 4096 + bcol], 0, 0);
      }
      #pragma unroll
      for (int i = 0; i < 16; ++i) {
        int kk = (i << 1) + bkk;
        Bcur[kk * BS + bcc] = W[(size_t)(kl0 + kk) * 4096 + bcol];
      }
    }
#endif

    // stage activations (16 rows x 32 K) -- shared by all 8 waves
    #pragma unroll
    for (int i = 0; i < 2; ++i) {
      int e = (i << 8) + tid;
      int m = e >> 5, kk = e & 31;
      int n = mbase + m;
      int kidx = kl0 + kk;
      float v;
      if (seg == 0)      v = x[((size_t)n * T_ + t) * D_ + kidx];
      else if (seg == 1) v = hprev[(size_t)n * H_ + kidx];
      else               v = attn[(size_t)n * H_ + kidx];
      act_sh[m * AS + kk] = v;
    }

#if USE_TDM
    if (w == 0) {
      if (kc + 1 < 96) wait_tensorcnt_le(1);  // chunk kc has landed in LDS
      else             wait_tensorcnt_le(0);
    }
#endif
    __syncthreads();

    #pragma unroll
    for (int kk0 = 0; kk0 < 32; kk0 += 4) {
      v2f af, bf;
      af[0] = act_sh[ar * AS + kk0 + khalf];
      af[1] = act_sh[ar * AS + kk0 + khalf + 1];
      bf[0] = Bcur[(kk0 + khalf) * BS + colb + ar];
      bf[1] = Bcur[(kk0 + khalf + 1) * BS + colb + ar];
      acc = __builtin_amdgcn_wmma_f32_16x16x4_f32(false, af, false, bf,
                                                  (short)0, acc, false, false);
    }
  }
  __syncthreads();

  // spill accumulator tile to LDS (D layout: vgpr v -> row v + 8*(lane>=16))
  #pragma unroll
  for (int v = 0; v < 8; ++v) {
    int r = v + ((lane >> 4) << 3);
    a_sh[r * OS + colb + ar] = acc[v];
  }
  __syncthreads();

  // fused LSTM gate update: i/f/o/g for (n,h) all live in this block
  #pragma unroll
  for (int i = 0; i < 2; ++i) {
    int e = (i << 8) + tid;   // 0..511
    int r = e >> 5, c = e & 31;
    int n = mbase + r;
    int hcol = cg * 32 + c;
    float ai  = a_sh[r * OS +       c] + bias[hcol];
    float afv = a_sh[r * OS +  32 + c] + bias[1024 + hcol];
    float ao  = a_sh[r * OS +  64 + c] + bias[2048 + hcol];
    float ag  = a_sh[r * OS +  96 + c] + bias[3072 + hcol];
    float ig = 1.f / (1.f + __expf(-ai));
    float fg = 1.f / (1.f + __expf(-afv));
    float og = 1.f / (1.f + __expf(-ao));
    float gg = tanhf(ag);
    size_t idx = (size_t)n * H_ + hcol;
    float cn = fg * cstate[idx] + ig * gg;
    cstate[idx] = cn;
    float hv = og * tanhf(cn);
    hnext[idx] = hv;
    out[((size_t)n * T_ + t) * H_ + hcol] = hv;
  }
}

// ---------------------------------------------------------------------------
extern "C" void kernel_launch(void* const* d_in, const int* in_sizes, int n_in,
                              void* d_out, int out_size, void* d_ws, size_t ws_size,
                              hipStream_t stream) {
  (void)in_sizes; (void)n_in; (void)out_size; (void)ws_size;
  const float* x     = (const float*)d_in[0];
  const float* A     = (const float*)d_in[1];
  const float* Wx    = (const float*)d_in[2];
  const float* Wh    = (const float*)d_in[3];
  const float* Wattn = (const float*)d_in[4];
  const float* bias  = (const float*)d_in[5];
  float* out = (float*)d_out;
  float* ws  = (float*)d_ws;

  const size_t NH = (size_t)N_ * H_;
  float* hbuf0 = ws;
  float* hbuf1 = ws + NH;
  float* cst   = ws + 2 * NH;
  float* attn  = ws + 3 * NH;   // needs 4*NH floats = 2 MB of workspace

  init_pool_kernel<<<(N_ * H_) / 64, 256, 0, stream>>>(A, hbuf0, cst);
  for (int t = 0; t < T_; ++t) {
    const float* hp = (t & 1) ? hbuf1 : hbuf0;
    float*       hn = (t & 1) ? hbuf0 : hbuf1;
    attn_kernel<<<N_, 256, 0, stream>>>(A, hp, attn);
    gates_kernel<<<256, 256, 0, stream>>>(x, Wx, Wh, Wattn, bias, attn,
                                          hp, hn, cst, out, t);
  }
}